// EqvSelfAttention_50955492000421
// MI455X (gfx1250) — compile-verified
//
#include <hip/hip_runtime.h>
#include <stdint.h>

typedef __attribute__((ext_vector_type(2))) float v2f;
typedef __attribute__((ext_vector_type(8))) float v8f;

#define Bsz 8
#define Nn  1024
#define Vv  512
#define Hh  8
#define Dd  64
#define INFC 1.0e38f
#define KPAD 4
#define LROW (Vv + KPAD)   // 516 floats: rows land on distinct LDS banks

static __device__ __forceinline__ v8f wmma_f32(v2f a, v2f b, v8f c) {
  // D = A(16x4,f32) * B(4x16,f32) + C(16x16,f32)
  return __builtin_amdgcn_wmma_f32_16x16x4_f32(false, a, false, b, (short)0, c, false, false);
}

// ASYNCcnt-tracked DMA: 16 bytes global -> LDS, no VGPR staging.
static __device__ __forceinline__ void async_ld_b128(uint32_t lds_off, const float* gptr) {
  asm volatile("global_load_async_to_lds_b128 %0, %1, off"
               :: "v"(lds_off), "v"((uint64_t)(uintptr_t)gptr)
               : "memory");
}
static __device__ __forceinline__ void wait_async16() {
  asm volatile("s_wait_asynccnt 16" ::: "memory");   // prev tile (16 loads) done
}
static __device__ __forceinline__ void wait_async0() {
  asm volatile("s_wait_asynccnt 0" ::: "memory");
}

// ---------------------------------------------------------------------------
// Kernel 1: Q/K/V projections.  out = Y(8192x512) @ W(512x512) + bias
// One wave per 16x64 output tile (A fragments reused across 4 col-tiles).
// ---------------------------------------------------------------------------
__global__ __launch_bounds__(256) void qkv_gemm_kernel(
    const float* __restrict__ Y,
    const float* __restrict__ Wq, const float* __restrict__ bq,
    const float* __restrict__ Wk, const float* __restrict__ bk,
    const float* __restrict__ Wv, const float* __restrict__ bv,
    float* __restrict__ Qo, float* __restrict__ Ko, float* __restrict__ Vo)
{
  const float* W; const float* bias; float* Out;
  if (blockIdx.y == 0)      { W = Wq; bias = bq; Out = Qo; }
  else if (blockIdx.y == 1) { W = Wk; bias = bk; Out = Ko; }
  else                      { W = Wv; bias = bv; Out = Vo; }

  const int lane = threadIdx.x & 31;
  const int wave = threadIdx.x >> 5;
  const int tile = blockIdx.x * 8 + wave;   // 512 row-tiles x 8 col-groups
  const int r0 = (tile >> 3) * 16;
  const int cg = (tile & 7) * 64;
  const int m  = lane & 15;
  const int kh = lane >> 4;

  v8f acc0 = {}, acc1 = {}, acc2 = {}, acc3 = {};
  const float* Ap0 = Y + (size_t)(r0 + m) * Vv + 2 * kh;
  const float* Bp0 = W + (size_t)(2 * kh) * Vv + cg + m;
#pragma unroll 4
  for (int k0 = 0; k0 < Vv; k0 += 4) {
    v2f a;
    a.x = Ap0[k0]; a.y = Ap0[k0 + 1];
    const float* Bp = Bp0 + (size_t)k0 * Vv;
    v2f b0, b1, b2, b3;
    b0.x = Bp[ 0]; b0.y = Bp[Vv +  0];
    b1.x = Bp[16]; b1.y = Bp[Vv + 16];
    b2.x = Bp[32]; b2.y = Bp[Vv + 32];
    b3.x = Bp[48]; b3.y = Bp[Vv + 48];
    acc0 = wmma_f32(a, b0, acc0);
    acc1 = wmma_f32(a, b1, acc1);
    acc2 = wmma_f32(a, b2, acc2);
    acc3 = wmma_f32(a, b3, acc3);
  }
  const float bb0 = bias[cg + m +  0];
  const float bb1 = bias[cg + m + 16];
  const float bb2 = bias[cg + m + 32];
  const float bb3 = bias[cg + m + 48];
  float* Op = Out + (size_t)(r0 + 8 * kh) * Vv + cg + m;
#pragma unroll
  for (int j = 0; j < 8; ++j) {
    Op[(size_t)j * Vv +  0] = acc0[j] + bb0;
    Op[(size_t)j * Vv + 16] = acc1[j] + bb1;
    Op[(size_t)j * Vv + 32] = acc2[j] + bb2;
    Op[(size_t)j * Vv + 48] = acc3[j] + bb3;
  }
}

// ---------------------------------------------------------------------------
// Kernel 2: fused flash-attention over keys.
// Block = (b, 16 query rows); wave w = head w.  Double-buffered: iteration i
// DMAs tile i+1's 16x512 K/V panels to LDS (global_load_async_to_lds_b128)
// while tile i's WMMAs/softmax run; in-order ASYNCcnt retirement means
// "s_wait_asynccnt 16" proves tile i has landed.
// ---------------------------------------------------------------------------
__global__ __launch_bounds__(256) void attn_kernel(
    const float* __restrict__ Qw, const float* __restrict__ Kw,
    const float* __restrict__ Vw, const float* __restrict__ Xp,
    const float* __restrict__ pq, const float* __restrict__ pk,
    const float* __restrict__ Wx1, const float* __restrict__ bx1,
    const float* __restrict__ Wx2, const float* __restrict__ bx2,
    float* __restrict__ Ow)
{
  __shared__ float lds_k[2][16][LROW];
  __shared__ float lds_v[2][16][LROW];
  __shared__ float p_s[8][16][17];     // wave-private: no block barrier needed
  __shared__ float loc_s[16][16];
  __shared__ float pq_s[16];

  const int b    = blockIdx.x >> 6;
  const int n0   = (blockIdx.x & 63) * 16;
  const int tid  = threadIdx.x;
  const int lane = tid & 31;
  const int h    = tid >> 5;          // wave == head
  const int m16  = lane & 15;
  const int kh   = lane >> 4;
  const float scale = 0.044194173824159216f;   // 1/sqrt(V)

  // DMA work split: 256 threads x 8 chunks x 16B cover one 16x512 f32 tile.
  const int t7 = tid >> 7;          // row parity
  const int tc = tid & 127;         // 16B-chunk within row
  const uint32_t kbase_l[2] = { (uint32_t)(uintptr_t)&lds_k[0][0][0],
                                (uint32_t)(uintptr_t)&lds_k[1][0][0] };
  const uint32_t vbase_l[2] = { (uint32_t)(uintptr_t)&lds_v[0][0][0],
                                (uint32_t)(uintptr_t)&lds_v[1][0][0] };

  // ---- prefetch tile 0 before anything else ----
  {
    const size_t grow = (size_t)(b * Nn) * Vv + tc * 4;
#pragma unroll
    for (int i = 0; i < 8; ++i) {
      const int row = 2 * i + t7;
      const uint32_t loff = (uint32_t)((row * LROW + tc * 4) * 4);
      async_ld_b128(kbase_l[0] + loff, Kw + grow + (size_t)row * Vv);
      async_ld_b128(vbase_l[0] + loff, Vw + grow + (size_t)row * Vv);
    }
  }

  // loc-MLP weights in registers (tiny)
  float wx1[9], bx1r[3], wx2r[3];
#pragma unroll
  for (int i = 0; i < 9; ++i) wx1[i] = Wx1[i];
#pragma unroll
  for (int i = 0; i < 3; ++i) { bx1r[i] = bx1[i]; wx2r[i] = Wx2[i]; }
  const float bx2r = bx2[0];

  if (tid < 16) pq_s[tid] = pq[b * Nn + n0 + tid];

  // Q fragments in A-layout: lane (kh*16+m) holds Q[n0+m][k0+2kh .. +1]
  v2f aq[16];
  const float* Qbase = Qw + ((size_t)(b * Nn + n0 + m16)) * Vv + h * Dd;
#pragma unroll
  for (int ks = 0; ks < 16; ++ks) {
    aq[ks].x = Qbase[ks * 4 + 2 * kh];
    aq[ks].y = Qbase[ks * 4 + 2 * kh + 1];
  }

  v8f o0 = {}, o1 = {}, o2 = {}, o3 = {};
  float mrow[8], lrow[8];
#pragma unroll
  for (int j = 0; j < 8; ++j) { mrow[j] = -3.0e38f; lrow[j] = 0.0f; }

  const int r_loc = tid >> 4, c_loc = tid & 15;

  __syncthreads();

  for (int it = 0; it < Nn / 16; ++it) {
    const int m0  = it * 16;
    const int cur = it & 1;
    const bool more = (it + 1) < (Nn / 16);

    // ---- kick off NEXT tile's K/V DMA (buffers freed by last iter's sync) --
    if (more) {
      const size_t grow = (size_t)(b * Nn + m0 + 16) * Vv + tc * 4;
#pragma unroll
      for (int i = 0; i < 8; ++i) {
        const int row = 2 * i + t7;
        const uint32_t loff = (uint32_t)((row * LROW + tc * 4) * 4);
        async_ld_b128(kbase_l[1 - cur] + loff, Kw + grow + (size_t)row * Vv);
        async_ld_b128(vbase_l[1 - cur] + loff, Vw + grow + (size_t)row * Vv);
      }
    }

    // ---- loc(b, n0+r, m0+c) MLP, shared by all heads (overlaps DMA) ----
    {
      const float* xp = Xp + (((size_t)(b * Nn + n0 + r_loc)) * Nn + (m0 + c_loc)) * 3;
      const float x0 = xp[0], x1 = xp[1], x2 = xp[2];
      float acc = bx2r;
#pragma unroll
      for (int i = 0; i < 3; ++i) {
        float hi = x0 * wx1[i] + x1 * wx1[3 + i] + x2 * wx1[6 + i] + bx1r[i];
        hi = fmaxf(hi, 0.0f);
        acc += hi * wx2r[i];
      }
      loc_s[r_loc][c_loc] = acc;
    }
    const float pkv = pk[b * Nn + m0 + m16];

    if (more) wait_async16(); else wait_async0();   // current tile landed
    __syncthreads();                                // lds_k/v[cur], loc_s visible

    // ---- S = Qh @ Kh^T from LDS (16x16, K-dim = 64 -> 16 WMMAs) ----
    v8f s = {};
#pragma unroll
    for (int ks = 0; ks < 16; ++ks) {
      v2f bf;
      bf.x = lds_k[cur][m16][h * Dd + ks * 4 + 2 * kh];
      bf.y = lds_k[cur][m16][h * Dd + ks * 4 + 2 * kh + 1];
      s = wmma_f32(aq[ks], bf, s);
    }

    // ---- masked online softmax in C-layout (row r = j+8*kh, col = m16) ----
    float alpha[8];
#pragma unroll
    for (int j = 0; j < 8; ++j) {
      const int r = j + 8 * kh;
      float val = s[j] * scale + loc_s[r][m16];
      const float pqv = pq_s[r];
      val = pqv * val - (1.0f - pqv) * INFC;
      val = pkv * val - (1.0f - pkv) * INFC;
      float vm = val;
#pragma unroll
      for (int off = 1; off < 16; off <<= 1) vm = fmaxf(vm, __shfl_xor(vm, off, 32));
      const float mo = mrow[j];
      const float mn = fmaxf(mo, vm);
      const float p  = __expf(val - mn);
      float psum = p;
#pragma unroll
      for (int off = 1; off < 16; off <<= 1) psum += __shfl_xor(psum, off, 32);
      alpha[j] = __expf(mo - mn);
      lrow[j]  = lrow[j] * alpha[j] + psum;
      mrow[j]  = mn;
      p_s[h][r][m16] = p;            // wave-private tile; LDS is in-order
    }
#pragma unroll
    for (int j = 0; j < 8; ++j) {
      o0[j] *= alpha[j]; o1[j] *= alpha[j]; o2[j] *= alpha[j]; o3[j] *= alpha[j];
    }

    // ---- O += P @ Vh from LDS (16x64, K=16 -> 4 k-steps x 4 d-tiles) ----
#pragma unroll
    for (int ks = 0; ks < 4; ++ks) {
      v2f ap;
      ap.x = p_s[h][m16][ks * 4 + 2 * kh];
      ap.y = p_s[h][m16][ks * 4 + 2 * kh + 1];
      const int row = ks * 4 + 2 * kh;
      v2f b0, b1, b2, b3;
      b0.x = lds_v[cur][row][h * Dd + m16 +  0]; b0.y = lds_v[cur][row + 1][h * Dd + m16 +  0];
      b1.x = lds_v[cur][row][h * Dd + m16 + 16]; b1.y = lds_v[cur][row + 1][h * Dd + m16 + 16];
      b2.x = lds_v[cur][row][h * Dd + m16 + 32]; b2.y = lds_v[cur][row + 1][h * Dd + m16 + 32];
      b3.x = lds_v[cur][row][h * Dd + m16 + 48]; b3.y = lds_v[cur][row + 1][h * Dd + m16 + 48];
      o0 = wmma_f32(ap, b0, o0);
      o1 = wmma_f32(ap, b1, o1);
      o2 = wmma_f32(ap, b2, o2);
      o3 = wmma_f32(ap, b3, o3);
    }
    __syncthreads();   // all reads of buf[cur]/loc_s done -> reusable next iter
  }

  // ---- normalize, add Qh residual, store Oh ----
#pragma unroll
  for (int j = 0; j < 8; ++j) {
    const int r = j + 8 * kh;
    const float inv = 1.0f / lrow[j];
    const size_t base = ((size_t)(b * Nn + n0 + r)) * Vv + h * Dd + m16;
    Ow[base +  0] = Qw[base +  0] + o0[j] * inv;
    Ow[base + 16] = Qw[base + 16] + o1[j] * inv;
    Ow[base + 32] = Qw[base + 32] + o2[j] * inv;
    Ow[base + 48] = Qw[base + 48] + o3[j] * inv;
  }
}

// ---------------------------------------------------------------------------
// Kernel 3: out = O + relu(O @ Wo + bo)   (16x64 tile per wave)
// ---------------------------------------------------------------------------
__global__ __launch_bounds__(256) void out_gemm_kernel(
    const float* __restrict__ O, const float* __restrict__ Wo,
    const float* __restrict__ bo, float* __restrict__ out)
{
  const int lane = threadIdx.x & 31;
  const int wave = threadIdx.x >> 5;
  const int tile = blockIdx.x * 8 + wave;
  const int r0 = (tile >> 3) * 16;
  const int cg = (tile & 7) * 64;
  const int m  = lane & 15;
  const int kh = lane >> 4;

  v8f acc0 = {}, acc1 = {}, acc2 = {}, acc3 = {};
  const float* Ap0 = O  + (size_t)(r0 + m) * Vv + 2 * kh;
  const float* Bp0 = Wo + (size_t)(2 * kh) * Vv + cg + m;
#pragma unroll 4
  for (int k0 = 0; k0 < Vv; k0 += 4) {
    v2f a;
    a.x = Ap0[k0]; a.y = Ap0[k0 + 1];
    const float* Bp = Bp0 + (size_t)k0 * Vv;
    v2f b0, b1, b2, b3;
    b0.x = Bp[ 0]; b0.y = Bp[Vv +  0];
    b1.x = Bp[16]; b1.y = Bp[Vv + 16];
    b2.x = Bp[32]; b2.y = Bp[Vv + 32];
    b3.x = Bp[48]; b3.y = Bp[Vv + 48];
    acc0 = wmma_f32(a, b0, acc0);
    acc1 = wmma_f32(a, b1, acc1);
    acc2 = wmma_f32(a, b2, acc2);
    acc3 = wmma_f32(a, b3, acc3);
  }
  const float bb0 = bo[cg + m +  0];
  const float bb1 = bo[cg + m + 16];
  const float bb2 = bo[cg + m + 32];
  const float bb3 = bo[cg + m + 48];
  const size_t row0 = (size_t)(r0 + 8 * kh);
#pragma unroll
  for (int j = 0; j < 8; ++j) {
    const size_t idx = (row0 + j) * Vv + cg + m;
    out[idx +  0] = O[idx +  0] + fmaxf(acc0[j] + bb0, 0.0f);
    out[idx + 16] = O[idx + 16] + fmaxf(acc1[j] + bb1, 0.0f);
    out[idx + 32] = O[idx + 32] + fmaxf(acc2[j] + bb2, 0.0f);
    out[idx + 48] = O[idx + 48] + fmaxf(acc3[j] + bb3, 0.0f);
  }
}

// ---------------------------------------------------------------------------
extern "C" void kernel_launch(void* const* d_in, const int* in_sizes, int n_in,
                              void* d_out, int out_size, void* d_ws, size_t ws_size,
                              hipStream_t stream) {
  (void)in_sizes; (void)n_in; (void)out_size; (void)ws_size;
  const float* Y   = (const float*)d_in[1];   // d_in[0] = X_lift (unused by ref)
  const float* Xp  = (const float*)d_in[2];
  const float* pq  = (const float*)d_in[3];
  const float* pk  = (const float*)d_in[4];
  const float* Wq  = (const float*)d_in[5];
  const float* bq  = (const float*)d_in[6];
  const float* Wk  = (const float*)d_in[7];
  const float* bk  = (const float*)d_in[8];
  const float* Wv  = (const float*)d_in[9];
  const float* bv  = (const float*)d_in[10];
  const float* Wo  = (const float*)d_in[11];
  const float* bo  = (const float*)d_in[12];
  const float* Wx1 = (const float*)d_in[13];
  const float* bx1 = (const float*)d_in[14];
  const float* Wx2 = (const float*)d_in[15];
  const float* bx2 = (const float*)d_in[16];
  float* out = (float*)d_out;

  const size_t mat = (size_t)Bsz * Nn * Vv;     // 4M floats = 16 MB
  float* Qw = (float*)d_ws;
  float* Kw = Qw + mat;
  float* Vw = Kw + mat;
  float* Ow = Vw + mat;                         // total 64 MB of d_ws

  qkv_gemm_kernel<<<dim3(512, 3), dim3(256), 0, stream>>>(
      Y, Wq, bq, Wk, bk, Wv, bv, Qw, Kw, Vw);
  attn_kernel<<<dim3(Bsz * (Nn / 16)), dim3(256), 0, stream>>>(
      Qw, Kw, Vw, Xp, pq, pk, Wx1, bx1, Wx2, bx2, Ow);
  out_gemm_kernel<<<dim3(512), dim3(256), 0, stream>>>(Ow, Wo, bo, out);
}